// AdaConv2d_996432413454
// MI455X (gfx1250) — compile-verified
//
#include <hip/hip_runtime.h>
#include <cstdint>

// AdaConv2d: out[b,c,h,w] = sum_{i,j in 0..2} x_pad[b,c,h+i-1,w+j-1] * dk[b,c,3i+j,h,w]
// fp32, B=8, C=64, H=W=128. Memory-bound (AI ~0.41 FLOP/B): optimize data movement.
//  - dk streamed once with non-temporal b128 loads (protect L2 residency of x)
//  - x staged to LDS via CDNA5 global_load_async_to_lds_b128 (ASYNCcnt) + s_wait_asynccnt
//  - out written with non-temporal b128 stores

typedef float v4f __attribute__((ext_vector_type(4)));

#define B_ 8
#define C_ 64
#define H_ 128
#define W_ 128
#define TR 8            // output rows per block
#define LROWS (TR + 2)  // staged rows incl. top/bottom halo

__global__ __launch_bounds__(256) void adaconv_dyn3x3_kernel(
    const float* __restrict__ x, const float* __restrict__ dk,
    float* __restrict__ out) {
  __shared__ __align__(16) float tile[LROWS * W_];  // 10 x 128 floats = 5 KB

  const int tid = threadIdx.x;
  const int tx = tid & 31;   // col group: 32 float4 = 128 cols (one wave per row)
  const int ty = tid >> 5;   // output row within tile, 0..7
  const int h0 = blockIdx.x * TR;
  const int c = blockIdx.y;
  const int b = blockIdx.z;

  const float* xbc = x + (size_t)(b * C_ + c) * (H_ * W_);

  // ---- Stage x rows [h0-1 .. h0+TR] into LDS with async DMA ----
  for (int i = tid; i < LROWS * 32; i += 256) {
    const int r = i >> 5;
    const int c4 = (i & 31) * 4;
    const int xrow = h0 - 1 + r;
    float* ldst = &tile[r * W_ + c4];
    if ((unsigned)xrow < (unsigned)H_) {
      const uint32_t lds_addr = (uint32_t)(uintptr_t)ldst;  // 0-based LDS byte offset
      const float* gsrc = xbc + (size_t)xrow * W_ + c4;
      asm volatile("global_load_async_to_lds_b128 %0, %1, off"
                   :: "v"(lds_addr), "v"(gsrc)
                   : "memory");
    } else {
      *(v4f*)ldst = (v4f){0.0f, 0.0f, 0.0f, 0.0f};  // zero-pad rows h=-1 / h=H
    }
  }
  asm volatile("s_wait_asynccnt 0x0" ::: "memory");
  __syncthreads();

  // ---- Compute 4 outputs per thread ----
  const int h = h0 + ty;
  const int w0 = tx * 4;
  const float* dkbc = dk + (size_t)(b * C_ + c) * 9 * (H_ * W_);

  v4f acc = {0.0f, 0.0f, 0.0f, 0.0f};
#pragma unroll
  for (int di = 0; di < 3; ++di) {
    const float* lrow = &tile[(ty + di) * W_];  // x row (h-1+di)
    float xr6[6];  // x cols w0-1 .. w0+4 (zero outside image)
    xr6[0] = (w0 > 0) ? lrow[w0 - 1] : 0.0f;
    const v4f m = *(const v4f*)&lrow[w0];
    xr6[1] = m.x; xr6[2] = m.y; xr6[3] = m.z; xr6[4] = m.w;
    xr6[5] = (w0 < W_ - 4) ? lrow[w0 + 4] : 0.0f;
#pragma unroll
    for (int dj = 0; dj < 3; ++dj) {
      const int tap = di * 3 + dj;
      const v4f kv = __builtin_nontemporal_load(
          (const v4f*)(dkbc + ((size_t)tap * H_ + h) * W_ + w0));
      acc.x += xr6[dj + 0] * kv.x;
      acc.y += xr6[dj + 1] * kv.y;
      acc.z += xr6[dj + 2] * kv.z;
      acc.w += xr6[dj + 3] * kv.w;
    }
  }
  __builtin_nontemporal_store(
      acc, (v4f*)(out + ((size_t)(b * C_ + c) * H_ + h) * W_ + w0));
}

extern "C" void kernel_launch(void* const* d_in, const int* in_sizes, int n_in,
                              void* d_out, int out_size, void* d_ws, size_t ws_size,
                              hipStream_t stream) {
  const float* x = (const float*)d_in[0];        // [8,64,128,128] fp32
  const float* dk = (const float*)d_in[1];       // [8,64,9,128,128] fp32
  float* out = (float*)d_out;                    // [8,64,128,128] fp32
  (void)in_sizes; (void)n_in; (void)out_size; (void)d_ws; (void)ws_size;

  dim3 grid(H_ / TR, C_, B_);  // (16, 64, 8) blocks, 256 threads (8 wave32) each
  adaconv_dyn3x3_kernel<<<grid, dim3(256), 0, stream>>>(x, dk, out);
}